// TreeMLPEncoder_10170482557586
// MI455X (gfx1250) — compile-verified
//
#include <hip/hip_runtime.h>

#define DD 256
#define NSTRUCT 8192
#define EPS 1e-5f

typedef float v2f __attribute__((ext_vector_type(2)));
typedef float v8f __attribute__((ext_vector_type(8)));

__device__ __forceinline__ float wave_sum(float v) {
  v += __shfl_xor(v, 1, 32);
  v += __shfl_xor(v, 2, 32);
  v += __shfl_xor(v, 4, 32);
  v += __shfl_xor(v, 8, 32);
  v += __shfl_xor(v, 16, 32);
  return v;
}
__device__ __forceinline__ float sum4(float4 x) { return x.x + x.y + x.z + x.w; }
__device__ __forceinline__ float sumsq4(float4 x) { return x.x*x.x + x.y*x.y + x.z*x.z + x.w*x.w; }
__device__ __forceinline__ float4 ln4(float4 x, float mu, float rs, float4 g, float4 b) {
  float4 y;
  y.x = (x.x - mu) * rs * g.x + b.x;
  y.y = (x.y - mu) * rs * g.y + b.y;
  y.z = (x.z - mu) * rs * g.z + b.z;
  y.w = (x.w - mu) * rs * g.w + b.w;
  return y;
}

// ---------------- Leaf gather + LayerNorm: one wave per 256-float row ----------------
__global__ __launch_bounds__(256) void leaf_ln_kernel(
    const float* __restrict__ comp_emb, const int* __restrict__ leaf_ids,
    const float* __restrict__ gamma, const float* __restrict__ beta,
    float* __restrict__ h0) {
  const int lane = threadIdx.x & 31;
  const int r = blockIdx.x * 8 + (threadIdx.x >> 5);     // 65536 rows
  const int id = leaf_ids[r];
  const float4* src = (const float4*)(comp_emb + (size_t)id * DD);
  float4 x0 = src[lane];
  float4 x1 = src[lane + 32];
  float s  = wave_sum(sum4(x0) + sum4(x1));
  float s2 = wave_sum(sumsq4(x0) + sumsq4(x1));
  float mu = s * (1.0f / DD);
  float rs = rsqrtf(s2 * (1.0f / DD) - mu * mu + EPS);
  float4 g0 = ((const float4*)gamma)[lane], g1 = ((const float4*)gamma)[lane + 32];
  float4 b0 = ((const float4*)beta)[lane],  b1 = ((const float4*)beta)[lane + 32];
  float4* dst = (float4*)(h0 + (size_t)r * DD);
  dst[lane]      = ln4(x0, mu, rs, g0, b0);
  dst[lane + 32] = ln4(x1, mu, rs, g1, b1);
}

// ---------------- Binary combine level: [op | left | right](768) @ W(768x256) + b, LN ----------------
#define MT 32      // output rows per block
#define KC 768     // reduction dim
#define CSTR 772   // padded LDS stride for cat tile (floats)
#define OSTR 260   // padded LDS stride for C tile (floats)

__global__ __launch_bounds__(256) void combine_kernel(
    const float* __restrict__ h_in, float* __restrict__ h_out,
    const int* __restrict__ op_ids, int off, int ln2n,
    const float* __restrict__ op_emb, const float* __restrict__ W,
    const float* __restrict__ bias, const float* __restrict__ gamma,
    const float* __restrict__ beta) {
  __shared__ float smem[MT * CSTR];   // reused: cat tile (32x768), then C tile (32x256 padded)

  const int tid = threadIdx.x;
  const int lane = tid & 31;
  const int wave = tid >> 5;
  const int r0 = blockIdx.x * MT;
  const int nmask = (1 << ln2n) - 1;

  // Phase 1: gather cat rows into LDS (float4 granularity; 24 iters/thread)
  for (int i = tid; i < MT * (KC / 4); i += 256) {
    const int m = i / (KC / 4);
    const int c4 = i - m * (KC / 4);        // 0..191
    const int rg = r0 + m;
    float4 v;
    if (c4 < 64) {
      const int opid = op_ids[(rg >> ln2n) * 7 + off + (rg & nmask)];
      v = ((const float4*)(op_emb + (size_t)opid * DD))[c4];
    } else if (c4 < 128) {
      v = ((const float4*)(h_in + (size_t)(2 * rg) * DD))[c4 - 64];
    } else {
      v = ((const float4*)(h_in + (size_t)(2 * rg + 1) * DD))[c4 - 128];
    }
    *(float4*)&smem[m * CSTR + c4 * 4] = v;
  }
  __syncthreads();

  // Phase 2: fp32 WMMA GEMM, K=768 in steps of 4. Wave owns cols [n0,n0+32), rows split 0-15/16-31.
  v8f c00 = {}, c01 = {}, c10 = {}, c11 = {};
  const int n0 = wave * 32;
  const int colA = n0 + (lane & 15);
  const int krow = (lane & 16) >> 3;                 // lanes 16-31 hold K=k0+2,k0+3
  const float* arow0 = &smem[(lane & 15) * CSTR];
  const float* arow1 = &smem[((lane & 15) + 16) * CSTR];
#pragma unroll 4
  for (int k0 = 0; k0 < KC; k0 += 4) {
    const int kb = k0 + krow;
    const float* Wk = W + (size_t)kb * DD;
    v2f b0 = { Wk[colA],      Wk[DD + colA] };
    v2f b1 = { Wk[colA + 16], Wk[DD + colA + 16] };
    v2f a0 = *(const v2f*)(arow0 + kb);
    v2f a1 = *(const v2f*)(arow1 + kb);
    c00 = __builtin_amdgcn_wmma_f32_16x16x4_f32(false, a0, false, b0, (short)0, c00, false, false);
    c01 = __builtin_amdgcn_wmma_f32_16x16x4_f32(false, a0, false, b1, (short)0, c01, false, false);
    c10 = __builtin_amdgcn_wmma_f32_16x16x4_f32(false, a1, false, b0, (short)0, c10, false, false);
    c11 = __builtin_amdgcn_wmma_f32_16x16x4_f32(false, a1, false, b1, (short)0, c11, false, false);
  }
  __syncthreads();   // cat tile dead; reuse LDS for C tile

  // Phase 3: spill accumulators (+bias) to LDS. C layout: VGPR j -> row j (lanes<16) / j+8 (lanes>=16)
  const float bias0 = bias[colA];
  const float bias1 = bias[colA + 16];
  const int roff = (lane & 16) >> 1;  // 0 or 8
#pragma unroll
  for (int j = 0; j < 8; ++j) {
    const int rr = j + roff;
    smem[rr * OSTR + colA]              = c00[j] + bias0;
    smem[rr * OSTR + colA + 16]         = c01[j] + bias1;
    smem[(rr + 16) * OSTR + colA]       = c10[j] + bias0;
    smem[(rr + 16) * OSTR + colA + 16]  = c11[j] + bias1;
  }
  __syncthreads();

  // Phase 4: LayerNorm 4 rows per wave, write out
  const float4 g0  = ((const float4*)gamma)[lane];
  const float4 g1  = ((const float4*)gamma)[lane + 32];
  const float4 bt0 = ((const float4*)beta)[lane];
  const float4 bt1 = ((const float4*)beta)[lane + 32];
#pragma unroll
  for (int i = 0; i < 4; ++i) {
    const int row = wave * 4 + i;
    float4 x0 = *(const float4*)&smem[row * OSTR + lane * 4];
    float4 x1 = *(const float4*)&smem[row * OSTR + 128 + lane * 4];
    float s  = wave_sum(sum4(x0) + sum4(x1));
    float s2 = wave_sum(sumsq4(x0) + sumsq4(x1));
    float mu = s * (1.0f / DD);
    float rs = rsqrtf(s2 * (1.0f / DD) - mu * mu + EPS);
    float4* dst = (float4*)(h_out + (size_t)(r0 + row) * DD);
    dst[lane]      = ln4(x0, mu, rs, g0, bt0);
    dst[lane + 32] = ln4(x1, mu, rs, g1, bt1);
  }
}

// ---------------- Final masked gather: one wave per output row ----------------
__global__ __launch_bounds__(256) void gather_kernel(
    const float* __restrict__ root, const int* __restrict__ sidx,
    float* __restrict__ out) {
  const int lane = threadIdx.x & 31;
  const int r = blockIdx.x * 8 + (threadIdx.x >> 5);   // 65536 rows
  const int idx = sidx[r];
  float4* dst = (float4*)(out + (size_t)r * DD);
  if (idx > 1) {
    const float4* src = (const float4*)(root + (size_t)idx * DD);
    dst[lane]      = src[lane];
    dst[lane + 32] = src[lane + 32];
  } else {
    float4 z = make_float4(0.f, 0.f, 0.f, 0.f);
    dst[lane]      = z;
    dst[lane + 32] = z;
  }
}

extern "C" void kernel_launch(void* const* d_in, const int* in_sizes, int n_in,
                              void* d_out, int out_size, void* d_ws, size_t ws_size,
                              hipStream_t stream) {
  const int*   struct_indices = (const int*)d_in[0];
  const int*   leaf_ids       = (const int*)d_in[1];
  const int*   op_ids         = (const int*)d_in[2];
  const float* comp_emb       = (const float*)d_in[3];
  const float* op_emb         = (const float*)d_in[4];
  const float* W_bin          = (const float*)d_in[5];
  const float* b_bin          = (const float*)d_in[6];
  const float* gamma          = (const float*)d_in[7];
  const float* beta           = (const float*)d_in[8];

  // h0 (8192*8*256 fp32 = 64MB) lives in d_out (same size; fully overwritten by gather_kernel).
  float* h0   = (float*)d_out;
  float* h1   = (float*)d_ws;                          // 8192*4*256
  float* h2   = h1 + (size_t)NSTRUCT * 4 * DD;         // 8192*2*256
  float* root = h2 + (size_t)NSTRUCT * 2 * DD;         // 8192*256

  leaf_ln_kernel<<<NSTRUCT * 8 / 8, 256, 0, stream>>>(comp_emb, leaf_ids, gamma, beta, h0);
  combine_kernel<<<(NSTRUCT * 4) / MT, 256, 0, stream>>>(h0, h1, op_ids, 0, 2,
                                                         op_emb, W_bin, b_bin, gamma, beta);
  combine_kernel<<<(NSTRUCT * 2) / MT, 256, 0, stream>>>(h1, h2, op_ids, 4, 1,
                                                         op_emb, W_bin, b_bin, gamma, beta);
  combine_kernel<<<(NSTRUCT * 1) / MT, 256, 0, stream>>>(h2, root, op_ids, 6, 0,
                                                         op_emb, W_bin, b_bin, gamma, beta);
  gather_kernel<<<65536 / 8, 256, 0, stream>>>(root, struct_indices, (float*)d_out);
}